// TGNN_69346541962038
// MI455X (gfx1250) — compile-verified
//
#include <hip/hip_runtime.h>
#include <hip/hip_bf16.h>

typedef float v2f __attribute__((ext_vector_type(2)));
typedef float v8f __attribute__((ext_vector_type(8)));

#define NF 128
#define BN_EPS_F 1e-5f

// ---------- degree / normalization ----------
__global__ void k_fill1(float* p, int n) {
    int i = blockIdx.x * blockDim.x + threadIdx.x;
    if (i < n) p[i] = 1.0f;           // self-loop weight included up-front
}

__global__ void k_deg_accum(const long long* ei, const float* w, float* deg, int E) {
    int e = blockIdx.x * blockDim.x + threadIdx.x;
    if (e < E) unsafeAtomicAdd(&deg[(int)ei[E + e]], w[e]);
}

__global__ void k_rsqrt(float* p, int n) {
    int i = blockIdx.x * blockDim.x + threadIdx.x;
    if (i < n) p[i] = rsqrtf(p[i]);   // deg >= 1 always (self-loop)
}

__global__ void k_edge_norm(const long long* ei, const float* w, const float* dinv,
                            float* norm, int E) {
    int e = blockIdx.x * blockDim.x + threadIdx.x;
    if (e < E) norm[e] = dinv[(int)ei[e]] * w[e] * dinv[(int)ei[E + e]];
}

// ---------- WMMA GEMM: C[M,128] = concat_seg(A_seg)[M, nseg*128] @ W[nseg*128, 128] ----------
// One block = 8 waves; block owns a 16-row stripe, wave w owns columns [16w,16w+16).
// W segment (128x128 = 64KB) staged in LDS per k-segment.
__global__ __launch_bounds__(256) void k_gemm_wmma(
    const float* __restrict__ A0, const float* __restrict__ A1, const float* __restrict__ A2,
    const float* __restrict__ W, const float* __restrict__ bias, float* __restrict__ C,
    int M, int nseg, int fuse_bias_relu) {
    __shared__ float lw[NF * NF];   // 64 KB
    const int m0   = blockIdx.x * 16;
    const int wave = threadIdx.x >> 5;
    const int lane = threadIdx.x & 31;
    const int half = lane >> 4;     // 0: K pair {0,1}; 1: K pair {2,3}
    const int l    = lane & 15;
    const int n0   = wave * 16;

    v8f c = {};
    for (int seg = 0; seg < nseg; ++seg) {
        const float* Ws = W + (size_t)seg * NF * NF;
        for (int i = threadIdx.x; i < NF * NF; i += 256) lw[i] = Ws[i];
        __syncthreads();

        const float* A = (seg == 0) ? A0 : ((seg == 1) ? A1 : A2);
        int rA = m0 + l; if (rA >= M) rA = M - 1;
        const float* arow = A + (size_t)rA * NF;

#pragma unroll 4
        for (int kb = 0; kb < NF / 4; ++kb) {
            const int kk = kb * 4 + half * 2;
            v2f a = *(const v2f*)(arow + kk);           // A: rows m0+l, K = kk,kk+1
            v2f b;
            b.x = lw[kk * NF + n0 + l];                  // B: row kk,  col n0+l
            b.y = lw[(kk + 1) * NF + n0 + l];            // B: row kk+1
            c = __builtin_amdgcn_wmma_f32_16x16x4_f32(
                    false, a, false, b, (short)0, c, false, false);
        }
        __syncthreads();
    }

#pragma unroll
    for (int v = 0; v < 8; ++v) {
        const int row = m0 + v + half * 8;
        const int col = n0 + l;
        if (row < M) {
            float val = c[v];
            if (fuse_bias_relu) val = fmaxf(val + bias[col], 0.0f);
            C[(size_t)row * NF + col] = val;
        }
    }
}

// ---------- message passing ----------
__global__ void k_selfloop(const float* __restrict__ h, const float* __restrict__ dinv,
                           float* __restrict__ acc, long long total) {
    long long idx = (long long)blockIdx.x * blockDim.x + threadIdx.x;
    if (idx < total) {
        float d = dinv[(int)(idx >> 7)];
        acc[idx] = d * d * h[idx];
    }
}

// one wave per edge; lanes stride features so each atomic covers 128B contiguous
__global__ __launch_bounds__(256) void k_scatter(
    const long long* __restrict__ ei, const float* __restrict__ norm,
    const float* __restrict__ h, float* __restrict__ acc, int E) {
    int e = blockIdx.x * 8 + (threadIdx.x >> 5);
    if (e >= E) return;
    const int lane = threadIdx.x & 31;
    const int r = (int)ei[e];
    const int cc = (int)ei[E + e];
    const float nv = norm[e];
    const float* hs = h + (size_t)r * NF;
    float* ad = acc + (size_t)cc * NF;
#pragma unroll
    for (int j = 0; j < 4; ++j) {
        int f = j * 32 + lane;
        unsafeAtomicAdd(&ad[f], nv * hs[f]);
    }
}

// ---------- bias + relu + BN stats ----------
__global__ __launch_bounds__(128) void k_bias_relu_stats(
    float* __restrict__ acc, const float* __restrict__ bias,
    float* __restrict__ stats, int N) {
    const int f = threadIdx.x;
    const int r0 = blockIdx.x * 64;
    const int r1 = min(r0 + 64, N);
    const float b = bias[f];
    float s = 0.f, sq = 0.f;
    for (int r = r0; r < r1; ++r) {
        float v = fmaxf(acc[(size_t)r * NF + f] + b, 0.f);
        acc[(size_t)r * NF + f] = v;
        s += v; sq += v * v;
    }
    unsafeAtomicAdd(&stats[f], s);
    unsafeAtomicAdd(&stats[NF + f], sq);
}

__global__ void k_bn_finalize(const float* __restrict__ gamma, const float* __restrict__ beta,
                              float* __restrict__ stats, int N) {
    const int f = threadIdx.x;
    const float invN = 1.0f / (float)N;
    const float mean = stats[f] * invN;
    const float var  = stats[NF + f] * invN - mean * mean;   // biased variance
    const float inv  = rsqrtf(var + BN_EPS_F);
    const float sc   = gamma[f] * inv;
    stats[2 * NF + f] = sc;
    stats[3 * NF + f] = beta[f] - mean * sc;
}

__global__ void k_bn_apply(const float* __restrict__ t, const float* __restrict__ stats,
                           float* __restrict__ out, long long total) {
    long long idx = (long long)blockIdx.x * blockDim.x + threadIdx.x;
    if (idx < total) {
        int f = (int)(idx & (NF - 1));
        out[idx] = t[idx] * stats[2 * NF + f] + stats[3 * NF + f];
    }
}

// ---------- fc2: per-node dot(z[n], w2) ----------
__global__ __launch_bounds__(256) void k_fc2(
    const float* __restrict__ z, const float* __restrict__ w2,
    const float* __restrict__ b2, float* __restrict__ out, int N) {
    int n = blockIdx.x * 8 + (threadIdx.x >> 5);
    if (n >= N) return;
    const int lane = threadIdx.x & 31;
    const float* zr = z + (size_t)n * NF;
    float s = 0.f;
#pragma unroll
    for (int j = 0; j < 4; ++j) {
        int f = j * 32 + lane;
        s += zr[f] * w2[f];
    }
#pragma unroll
    for (int off = 16; off > 0; off >>= 1)
        s += __shfl_xor(s, off, 32);
    if (lane == 0) out[n] = fmaxf(s + b2[0], 0.f);
}

static inline size_t pad64(size_t n) { return (n + 63) & ~(size_t)63; }

extern "C" void kernel_launch(void* const* d_in, const int* in_sizes, int n_in,
                              void* d_out, int out_size, void* d_ws, size_t ws_size,
                              hipStream_t stream) {
    const float*     x   = (const float*)d_in[0];
    const long long* ei  = (const long long*)d_in[1];   // int64 [2, E]
    const float*     ew  = (const float*)d_in[2];
    const float*     W1  = (const float*)d_in[3];
    const float*     b1  = (const float*)d_in[4];
    const float*     W2  = (const float*)d_in[5];
    const float*     b2c = (const float*)d_in[6];
    const float*     g1  = (const float*)d_in[7];
    const float*     be1 = (const float*)d_in[8];
    const float*     g2  = (const float*)d_in[9];
    const float*     be2 = (const float*)d_in[10];
    const float*     fW1 = (const float*)d_in[11];      // [384,128]
    const float*     fb1 = (const float*)d_in[12];
    const float*     fW2 = (const float*)d_in[13];      // [128,1]
    const float*     fb2 = (const float*)d_in[14];
    float* out = (float*)d_out;

    const int N = in_sizes[0] / NF;
    const int E = in_sizes[2];
    const long long NH = (long long)N * NF;

    // workspace layout (floats, 256B-aligned chunks)
    float* ws    = (float*)d_ws;
    float* dinv  = ws;                       // N   (degree, then rsqrt in place)
    float* norm  = dinv + pad64(N);          // E
    float* htmp  = norm + pad64(E);          // N*128  (GEMM out; reused as z)
    float* acc   = htmp + NH;                // N*128  (scatter accumulator / relu'd t)
    float* h1    = acc  + NH;                // N*128
    float* h2    = h1   + NH;                // N*128
    float* stats = h2   + NH;                // 512: sum | sumsq | scale | shift
    float* z     = htmp;

    const int T = 256;
    const dim3 ggemm((N + 15) / 16);
    const int gNH = (int)((NH + T - 1) / T);

    // ---- gcn_norm ----
    k_fill1    <<<(N + T - 1) / T, T, 0, stream>>>(dinv, N);
    k_deg_accum<<<(E + T - 1) / T, T, 0, stream>>>(ei, ew, dinv, E);
    k_rsqrt    <<<(N + T - 1) / T, T, 0, stream>>>(dinv, N);
    k_edge_norm<<<(E + T - 1) / T, T, 0, stream>>>(ei, ew, dinv, norm, E);

    // ---- conv1 ----
    k_gemm_wmma<<<ggemm, T, 0, stream>>>(x, nullptr, nullptr, W1, nullptr, htmp, N, 1, 0);
    k_selfloop <<<gNH, T, 0, stream>>>(htmp, dinv, acc, NH);
    k_scatter  <<<(E + 7) / 8, T, 0, stream>>>(ei, norm, htmp, acc, E);
    hipMemsetAsync(stats, 0, 2 * NF * sizeof(float), stream);
    k_bias_relu_stats<<<(N + 63) / 64, NF, 0, stream>>>(acc, b1, stats, N);
    k_bn_finalize    <<<1, NF, 0, stream>>>(g1, be1, stats, N);
    k_bn_apply       <<<gNH, T, 0, stream>>>(acc, stats, h1, NH);

    // ---- conv2 ----
    k_gemm_wmma<<<ggemm, T, 0, stream>>>(h1, nullptr, nullptr, W2, nullptr, htmp, N, 1, 0);
    k_selfloop <<<gNH, T, 0, stream>>>(htmp, dinv, acc, NH);
    k_scatter  <<<(E + 7) / 8, T, 0, stream>>>(ei, norm, htmp, acc, E);
    hipMemsetAsync(stats, 0, 2 * NF * sizeof(float), stream);
    k_bias_relu_stats<<<(N + 63) / 64, NF, 0, stream>>>(acc, b2c, stats, N);
    k_bn_finalize    <<<1, NF, 0, stream>>>(g2, be2, stats, N);
    k_bn_apply       <<<gNH, T, 0, stream>>>(acc, stats, h2, NH);

    // ---- head: z = relu([x|h1|h2] @ fc1_W + fc1_b) without materializing cat ----
    k_gemm_wmma<<<ggemm, T, 0, stream>>>(x, h1, h2, fW1, fb1, z, N, 3, 1);
    k_fc2      <<<(N + 7) / 8, T, 0, stream>>>(z, fW2, fb2, out, N);
}